// HeteroGNN_1099511628126
// MI455X (gfx1250) — compile-verified
//
#include <hip/hip_runtime.h>
#include <hip/hip_bf16.h>
#include <math.h>

// ---------------------------------------------------------------------------
// CDNA5 (gfx1250) fp32 WMMA GNN pipeline.
// D = A(16x4) x B(4x16) + C(16x16), fp32 throughout (matches reference).
// Conv kernel: M=32 edges per wave (2 accumulator sets) so each packed
// B-fragment global load feeds two v_wmma ops.
// ---------------------------------------------------------------------------

typedef __attribute__((ext_vector_type(2))) float v2f;
typedef __attribute__((ext_vector_type(8))) float v8f;

#define LDS_PAD 4  // pad columns to dodge bank conflicts; keeps 16B alignment

// ---- 16xK tile GEMM (used by projections) --------------------------------
__device__ __forceinline__ void wmma_gemm(const float* tl, int ldt,
                                          const float2* __restrict__ pw,
                                          int ktc, v8f acc[8], int lane) {
  const int nl = lane & 15;
  const int kh = lane >> 4;
  for (int kt = 0; kt < ktc; ++kt) {
    float2 av = *(const float2*)(tl + nl * ldt + kt * 4 + 2 * kh);
    v2f a;
    a.x = av.x;
    a.y = av.y;
#pragma unroll
    for (int nt = 0; nt < 8; ++nt) {
      float2 bv = pw[(kt * 8 + nt) * 32 + lane];
      v2f b;
      b.x = bv.x;
      b.y = bv.y;
      acc[nt] = __builtin_amdgcn_wmma_f32_16x16x4_f32(
          false, a, false, b, (short)0, acc[nt], false, false);
    }
  }
}

// ---- 32xK tile GEMM (conv): one B-frag load feeds two WMMAs --------------
__device__ __forceinline__ void wmma_gemm2(const float* tl, int ldt,
                                           const float2* __restrict__ pw,
                                           int ktc, v8f acc[2][8], int lane) {
  const int nl = lane & 15;
  const int kh = lane >> 4;
  for (int kt = 0; kt < ktc; ++kt) {
    float2 av0 = *(const float2*)(tl + nl * ldt + kt * 4 + 2 * kh);
    float2 av1 = *(const float2*)(tl + (nl + 16) * ldt + kt * 4 + 2 * kh);
    v2f a0, a1;
    a0.x = av0.x;
    a0.y = av0.y;
    a1.x = av1.x;
    a1.y = av1.y;
#pragma unroll
    for (int nt = 0; nt < 8; ++nt) {
      float2 bv = pw[(kt * 8 + nt) * 32 + lane];
      v2f b;
      b.x = bv.x;
      b.y = bv.y;
      acc[0][nt] = __builtin_amdgcn_wmma_f32_16x16x4_f32(
          false, a0, false, b, (short)0, acc[0][nt], false, false);
      acc[1][nt] = __builtin_amdgcn_wmma_f32_16x16x4_f32(
          false, a1, false, b, (short)0, acc[1][nt], false, false);
    }
  }
}

// Pack a row-major [K x 128] weight into B-fragment order:
// pw[((kt*8+nt)*32 + lane)] = { w[(4kt+2kh)*128 + nt*16+nl],
//                               w[(4kt+2kh+1)*128 + nt*16+nl] }
__global__ __launch_bounds__(256) void pack_kernel(const float* __restrict__ w,
                                                   float* __restrict__ pw,
                                                   int ktc) {
  int idx = blockIdx.x * blockDim.x + threadIdx.x;
  int total = ktc * 8 * 32;
  if (idx >= total) return;
  int lane = idx & 31;
  int nt = (idx >> 5) & 7;
  int kt = idx >> 8;
  int kh = lane >> 4, nl = lane & 15;
  int k0 = kt * 4 + 2 * kh;
  pw[2 * idx + 0] = w[(size_t)k0 * 128 + nt * 16 + nl];
  pw[2 * idx + 1] = w[(size_t)(k0 + 1) * 128 + nt * 16 + nl];
}

__global__ __launch_bounds__(256) void count_kernel(const int* __restrict__ dst,
                                                    float* __restrict__ cnt,
                                                    int E) {
  int i = blockIdx.x * blockDim.x + threadIdx.x;
  if (i < E) atomicAdd(&cnt[dst[i]], 1.0f);
}

// out = relu(sum / max(cnt,1))
__global__ __launch_bounds__(256) void finalize_kernel(
    const float* __restrict__ sum, const float* __restrict__ cnt,
    float* __restrict__ out, int total) {
  int i = blockIdx.x * blockDim.x + threadIdx.x;
  if (i >= total) return;
  float c = fmaxf(cnt[i >> 7], 1.0f);
  out[i] = fmaxf(sum[i] / c, 0.0f);
}

// ---------------------------------------------------------------------------
// Node projection: out[r,:] = nan_to_num(relu(x[r,:] @ W + b)), W packed.
// One wave = 16 rows. K = 64 or 128.
// ---------------------------------------------------------------------------
template <int K>
__global__ __launch_bounds__(128) void proj_kernel(
    const float* __restrict__ x, const float* __restrict__ pw,
    const float* __restrict__ bias, float* __restrict__ out, int nrows) {
  constexpr int LDT = K + LDS_PAD;
  __shared__ float lds[4 * 16 * LDT];
  const int lane = threadIdx.x & 31;
  const int wv = threadIdx.x >> 5;
  const int r0 = (blockIdx.x * 4 + wv) * 16;
  if (r0 >= nrows) return;
  float* tl = &lds[wv * 16 * LDT];
  const int nl = lane & 15, kh = lane >> 4;

  if (K == 64) {
#pragma unroll 4
    for (int r = 0; r < 16; ++r) {
      float2 v = *(const float2*)(x + (size_t)(r0 + r) * K + lane * 2);
      *(float2*)(tl + r * LDT + lane * 2) = v;
    }
  } else {
#pragma unroll 4
    for (int r = 0; r < 16; ++r) {
      float4 v = *(const float4*)(x + (size_t)(r0 + r) * K + lane * 4);
      *(float4*)(tl + r * LDT + lane * 4) = v;
    }
  }
  asm volatile("s_wait_dscnt 0" ::: "memory");

  v8f acc[8];
#pragma unroll
  for (int nt = 0; nt < 8; ++nt) {
    float b = bias[nt * 16 + nl];
#pragma unroll
    for (int r = 0; r < 8; ++r) acc[nt][r] = b;
  }
  wmma_gemm(tl, LDT, (const float2*)pw, K / 4, acc, lane);

#pragma unroll
  for (int nt = 0; nt < 8; ++nt) {
#pragma unroll
    for (int r = 0; r < 8; ++r) {
      float v = fmaxf(acc[nt][r], 0.0f);       // relu; v_max_num: NaN -> 0
      v = fminf(v, 3.402823466e+38f);          // +inf -> FLT_MAX (nan_to_num)
      out[(size_t)(r0 + r + 8 * kh) * 128 + nt * 16 + nl] = v;
    }
  }
}

// ---------------------------------------------------------------------------
// PRMPConv edge kernel. One wave = 32 edges (2 sets of 16):
//   h    = relu(x_dst[dst] @ w1 + b1)
//   pred = h @ w2 + b2
//   res  = x_src[src] - pred
//   msg  = LN(res; g,be) @ wl + bl
//   atomic-add msg into out_sum[dst]
// Tile row m = 16*s + r + 8*kh (set s, acc VGPR r, lane half kh).
// ---------------------------------------------------------------------------
__global__ __launch_bounds__(64) void prmp_conv_kernel(
    const float* __restrict__ xs, const float* __restrict__ xd,
    const int* __restrict__ sidxg, const int* __restrict__ didxg,
    const float* __restrict__ pw1, const float* __restrict__ b1,
    const float* __restrict__ pw2, const float* __restrict__ b2,
    const float* __restrict__ gg, const float* __restrict__ bb,
    const float* __restrict__ pwl, const float* __restrict__ bl,
    float* __restrict__ osum, int E) {
  constexpr int LDT = 128 + LDS_PAD;
  __shared__ float lds[2 * 32 * LDT];
  const int lane = threadIdx.x & 31;
  const int wv = threadIdx.x >> 5;
  const int e0 = (blockIdx.x * 2 + wv) * 32;
  if (e0 >= E) return;
  float* tl = &lds[wv * 32 * LDT];
  const int nl = lane & 15, kh = lane >> 4;

  // Edge ids in C-fragment row order.
  int sidx[2][8], didx[2][8];
#pragma unroll
  for (int s = 0; s < 2; ++s)
#pragma unroll
    for (int r = 0; r < 8; ++r) {
      int e = e0 + 16 * s + r + 8 * kh;
      sidx[s][r] = sidxg[e];
      didx[s][r] = didxg[e];
    }

  // Stage gathered x_dst tile (32 x 128) into LDS, 128B/row/lane-group.
#pragma unroll 4
  for (int r = 0; r < 32; ++r) {
    int d = didxg[e0 + r];
    float4 v = *(const float4*)(xd + (size_t)d * 128 + lane * 4);
    *(float4*)(tl + r * LDT + lane * 4) = v;
  }
  asm volatile("s_wait_dscnt 0" ::: "memory");

  v8f acc[2][8];
  // ---- GEMM1: h = relu(x_d @ w1 + b1) ----
#pragma unroll
  for (int nt = 0; nt < 8; ++nt) {
    float b = b1[nt * 16 + nl];
#pragma unroll
    for (int s = 0; s < 2; ++s)
#pragma unroll
      for (int r = 0; r < 8; ++r) acc[s][nt][r] = b;
  }
  wmma_gemm2(tl, LDT, (const float2*)pw1, 32, acc, lane);

  // h -> LDS (row-major)
#pragma unroll
  for (int s = 0; s < 2; ++s)
#pragma unroll
    for (int nt = 0; nt < 8; ++nt)
#pragma unroll
      for (int r = 0; r < 8; ++r)
        tl[(16 * s + r + 8 * kh) * LDT + nt * 16 + nl] =
            fmaxf(acc[s][nt][r], 0.0f);
  asm volatile("s_wait_dscnt 0" ::: "memory");

  // ---- GEMM2: pred = h @ w2 + b2 ----
#pragma unroll
  for (int nt = 0; nt < 8; ++nt) {
    float b = b2[nt * 16 + nl];
#pragma unroll
    for (int s = 0; s < 2; ++s)
#pragma unroll
      for (int r = 0; r < 8; ++r) acc[s][nt][r] = b;
  }
  wmma_gemm2(tl, LDT, (const float2*)pw2, 32, acc, lane);

  // ---- residual = x_j - pred (gather x_src directly in C layout) ----
#pragma unroll
  for (int s = 0; s < 2; ++s)
#pragma unroll
    for (int nt = 0; nt < 8; ++nt)
#pragma unroll
      for (int r = 0; r < 8; ++r) {
        float xj = xs[(size_t)sidx[s][r] * 128 + nt * 16 + nl];
        acc[s][nt][r] = xj - acc[s][nt][r];
      }

  // ---- LayerNorm per row (128 cols = 16 lanes x 8 regs) -> LDS ----
  float gv[8], bev[8];
#pragma unroll
  for (int nt = 0; nt < 8; ++nt) {
    gv[nt] = gg[nt * 16 + nl];
    bev[nt] = bb[nt * 16 + nl];
  }
#pragma unroll
  for (int s = 0; s < 2; ++s)
#pragma unroll
    for (int r = 0; r < 8; ++r) {
      float sm = 0.0f;
#pragma unroll
      for (int nt = 0; nt < 8; ++nt) sm += acc[s][nt][r];
      sm += __shfl_xor(sm, 1);
      sm += __shfl_xor(sm, 2);
      sm += __shfl_xor(sm, 4);
      sm += __shfl_xor(sm, 8);
      float mu = sm * (1.0f / 128.0f);
      float ss = 0.0f;
#pragma unroll
      for (int nt = 0; nt < 8; ++nt) {
        float d = acc[s][nt][r] - mu;
        ss += d * d;
      }
      ss += __shfl_xor(ss, 1);
      ss += __shfl_xor(ss, 2);
      ss += __shfl_xor(ss, 4);
      ss += __shfl_xor(ss, 8);
      float rs = rsqrtf(ss * (1.0f / 128.0f) + 1e-5f);
#pragma unroll
      for (int nt = 0; nt < 8; ++nt)
        tl[(16 * s + r + 8 * kh) * LDT + nt * 16 + nl] =
            (acc[s][nt][r] - mu) * rs * gv[nt] + bev[nt];
    }
  asm volatile("s_wait_dscnt 0" ::: "memory");

  // ---- GEMM3: msg = LN(res) @ wl + bl ----
#pragma unroll
  for (int nt = 0; nt < 8; ++nt) {
    float b = bl[nt * 16 + nl];
#pragma unroll
    for (int s = 0; s < 2; ++s)
#pragma unroll
      for (int r = 0; r < 8; ++r) acc[s][nt][r] = b;
  }
  wmma_gemm2(tl, LDT, (const float2*)pwl, 32, acc, lane);

  // ---- scatter-add into per-destination sums ----
#pragma unroll
  for (int s = 0; s < 2; ++s)
#pragma unroll
    for (int nt = 0; nt < 8; ++nt)
#pragma unroll
      for (int r = 0; r < 8; ++r)
        atomicAdd(osum + (size_t)didx[s][r] * 128 + nt * 16 + nl,
                  acc[s][nt][r]);
}

// ---------------------------------------------------------------------------
extern "C" void kernel_launch(void* const* d_in, const int* in_sizes, int n_in,
                              void* d_out, int out_size, void* d_ws,
                              size_t ws_size, hipStream_t stream) {
  const int NU = 100000, NI = 100000, E = 600000;
  const float* x_user = (const float*)d_in[0];
  const float* x_item = (const float*)d_in[1];
  const int* e_ui = (const int*)d_in[2];  // [2,E]: row0=src(user), row1=dst(item)
  const int* e_iu = (const int*)d_in[3];  // [2,E]: row0=src(item), row1=dst(user)
  const float* pu_w = (const float*)d_in[4];
  const float* pu_b = (const float*)d_in[5];
  const float* pi_w = (const float*)d_in[6];
  const float* pi_b = (const float*)d_in[7];
  auto cp = [&](int ci, int j) { return (const float*)d_in[8 + ci * 8 + j]; };
  // j: 0=w1 1=b1 2=w2 3=b2 4=g 5=be 6=wl 7=bl; ci: 0=c1_ui 1=c1_iu 2=c2_ui 3=c2_iu

  float* ws = (float*)d_ws;
  const size_t NB = (size_t)NU * 128;  // 12.8M floats per node buffer
  float* A = ws;            // hu / hu2
  float* Bf = ws + NB;      // hi / hi2
  float* C = ws + 2 * NB;   // user sums
  float* D = ws + 3 * NB;   // item sums
  float* cntU = ws + 4 * NB;
  float* cntI = cntU + NU;
  float* P = cntI + NI;
  float* pwU = P;                 // 16*8*64  = 8192 floats
  float* pwI = pwU + 8192;        // 32*8*64  = 16384
  float* pc = pwI + 16384;        // 12 x 16384

  // ---- pack all weight matrices into WMMA B-fragment order ----
  pack_kernel<<<(4096 + 255) / 256, 256, 0, stream>>>(pu_w, pwU, 16);
  pack_kernel<<<(8192 + 255) / 256, 256, 0, stream>>>(pi_w, pwI, 32);
  for (int ci = 0; ci < 4; ++ci) {
    pack_kernel<<<(8192 + 255) / 256, 256, 0, stream>>>(cp(ci, 0), pc + (ci * 3 + 0) * 16384, 32);
    pack_kernel<<<(8192 + 255) / 256, 256, 0, stream>>>(cp(ci, 2), pc + (ci * 3 + 1) * 16384, 32);
    pack_kernel<<<(8192 + 255) / 256, 256, 0, stream>>>(cp(ci, 6), pc + (ci * 3 + 2) * 16384, 32);
  }

  // ---- segment counts (same per edge list across both layers) ----
  hipMemsetAsync(cntU, 0, NU * sizeof(float), stream);
  hipMemsetAsync(cntI, 0, NI * sizeof(float), stream);
  count_kernel<<<(E + 255) / 256, 256, 0, stream>>>(e_ui + E, cntI, E);
  count_kernel<<<(E + 255) / 256, 256, 0, stream>>>(e_iu + E, cntU, E);

  // ---- projections ----
  proj_kernel<64><<<(NU / 16 + 3) / 4, 128, 0, stream>>>(x_user, pwU, pu_b, A, NU);
  proj_kernel<128><<<(NI / 16 + 3) / 4, 128, 0, stream>>>(x_item, pwI, pi_b, Bf, NI);

  const int cgrid = (E / 32 + 1) / 2;  // 32 edges per wave, 2 waves per block

  // ---- layer 1 ----
  hipMemsetAsync(C, 0, NB * sizeof(float), stream);
  hipMemsetAsync(D, 0, NB * sizeof(float), stream);
  prmp_conv_kernel<<<cgrid, 64, 0, stream>>>(
      A, Bf, e_ui, e_ui + E, pc + 0 * 16384, cp(0, 1), pc + 1 * 16384, cp(0, 3),
      cp(0, 4), cp(0, 5), pc + 2 * 16384, cp(0, 7), D, E);
  prmp_conv_kernel<<<cgrid, 64, 0, stream>>>(
      Bf, A, e_iu, e_iu + E, pc + 3 * 16384, cp(1, 1), pc + 4 * 16384, cp(1, 3),
      cp(1, 4), cp(1, 5), pc + 5 * 16384, cp(1, 7), C, E);
  finalize_kernel<<<((int)NB + 255) / 256, 256, 0, stream>>>(C, cntU, A, (int)NB);
  finalize_kernel<<<((int)NB + 255) / 256, 256, 0, stream>>>(D, cntI, Bf, (int)NB);

  // ---- layer 2 ----
  hipMemsetAsync(C, 0, NB * sizeof(float), stream);
  hipMemsetAsync(D, 0, NB * sizeof(float), stream);
  prmp_conv_kernel<<<cgrid, 64, 0, stream>>>(
      A, Bf, e_ui, e_ui + E, pc + 6 * 16384, cp(2, 1), pc + 7 * 16384, cp(2, 3),
      cp(2, 4), cp(2, 5), pc + 8 * 16384, cp(2, 7), D, E);
  prmp_conv_kernel<<<cgrid, 64, 0, stream>>>(
      Bf, A, e_iu, e_iu + E, pc + 9 * 16384, cp(3, 1), pc + 10 * 16384, cp(3, 3),
      cp(3, 4), cp(3, 5), pc + 11 * 16384, cp(3, 7), C, E);

  float* out = (float*)d_out;
  finalize_kernel<<<((int)NB + 255) / 256, 256, 0, stream>>>(C, cntU, out, (int)NB);
  finalize_kernel<<<((int)NB + 255) / 256, 256, 0, stream>>>(D, cntI, out + NB, (int)NB);
}